// ConduitHydrology_15384572854416
// MI455X (gfx1250) — compile-verified
//
#include <hip/hip_runtime.h>
#include <cstdint>

// ---------------------------------------------------------------------------
// Conduit hydrology residual on a fixed 1500x1500 raster (memory-bound stencil)
// MI455X strategy: one thread per node; CDNA5 async global->LDS staging of link
// data (b128 where 16B-aligned), n_eff halo rows staged via VALU+ds_store,
// hardware rcp/rsq instead of IEEE-div sequences to keep VALU << HBM time.
// ---------------------------------------------------------------------------

#define NROWS 1500
#define NCOLS 1500
#define NH (NROWS * (NCOLS - 1))   // number of horizontal links = 2,248,500
#define TILE 256

__device__ __forceinline__ uint32_t lds_off(const void* p) {
  // flat address of an LDS object: its low 32 bits are the LDS byte offset (ISA 10.2)
  return (uint32_t)(uintptr_t)p;
}

__device__ __forceinline__ void async_ld32(uint32_t lds, const float* base,
                                           uint32_t byte_off) {
  // GLOBAL_LOAD_ASYNC_TO_LDS_B32, GVS mode: copies 4 bytes from global base saddr plus byte offset voffset to the LDS address in vdst
  asm volatile("global_load_async_to_lds_b32 %0, %1, %2"
               :
               : "v"(lds), "v"(byte_off), "s"(base)
               : "memory");
}

__device__ __forceinline__ void async_ld128(uint32_t lds, const float* base,
                                            uint32_t byte_off) {
  // GLOBAL_LOAD_ASYNC_TO_LDS_B128: copies 16 bytes from global base saddr plus byte offset voffset to the LDS address in vdst
  asm volatile("global_load_async_to_lds_b128 %0, %1, %2"
               :
               : "v"(lds), "v"(byte_off), "s"(base)
               : "memory");
}

__device__ __forceinline__ float frcp(float x) { return __builtin_amdgcn_rcpf(x); }

__global__ __launch_bounds__(TILE) void ConduitHydrology_kernel(
    const float* __restrict__ eff,   // effective_pressure
    const float* __restrict__ dis,   // discharge
    const float* __restrict__ geo,   // geometric_gradient
    const float* __restrict__ ovb,   // overburden_pressure
    const float* __restrict__ sv,    // sliding_velocity (per link)
    const float* __restrict__ len,   // link_length (per link)
    const int*   __restrict__ status,
    float* __restrict__ out) {
  __shared__ float sN[3][TILE + 8];            // n_eff rows r-1,r,r+1; j -> col c0-1+j
  __shared__ float sH[TILE + 8];               // h-link sv, j -> link (r, c0-1+j)
  __shared__ alignas(16) float sU[TILE];       // v-link sv (r-1 -> r) at col c0+t
  __shared__ alignas(16) float sD[TILE];       // v-link sv (r -> r+1) at col c0+t

  const int r  = blockIdx.y;
  const int c0 = blockIdx.x * TILE;
  const int t  = threadIdx.x;

  // ---- stage n_eff (3 rows, with 1-col halo): needs the status select ----
  for (int slot = 0; slot < 3; ++slot) {
    const int rr = r - 1 + slot;
    if (rr < 0 || rr >= NROWS) continue;
    for (int j = t; j < TILE + 2; j += TILE) {
      const int cc = c0 - 1 + j;
      if (cc >= 0 && cc < NCOLS) {
        const int g = rr * NCOLS + cc;
        sN[slot][j] = (status[g] != 0) ? ovb[g] : eff[g];
      }
    }
  }

  // ---- stage sliding velocities via CDNA5 async global->LDS copies ----
  // horizontal halo row: base is off-by-one -> unaligned, use b32
  for (int j = t; j < TILE + 1; j += TILE) {
    const int cc = c0 - 1 + j;                 // h-link (r, cc)
    if (cc >= 0 && cc < NCOLS - 1) {
      async_ld32(lds_off(&sH[j]), sv, (uint32_t)(r * (NCOLS - 1) + cc) * 4u);
    }
  }
  // vertical link rows: (NH + k*NCOLS + c0)*4 is 16B-aligned and valid tile
  // widths (256 / 220) are multiples of 4 -> whole-chunk validity == first col.
  if (t < TILE / 4) {
    const int cq = c0 + 4 * t;                 // first column of this 16B chunk
    if (r > 0 && cq < NCOLS) {                 // v-links (r-1 -> r)
      async_ld128(lds_off(&sU[4 * t]), sv,
                  (uint32_t)(NH + (r - 1) * NCOLS + cq) * 4u);
    }
    if (r < NROWS - 1 && cq < NCOLS) {         // v-links (r -> r+1)
      async_ld128(lds_off(&sD[4 * t]), sv,
                  (uint32_t)(NH + r * NCOLS + cq) * 4u);
    }
  }
  asm volatile("s_wait_asynccnt 0" ::: "memory");
  __syncthreads();

  const int c = c0 + t;
  if (c >= NCOLS) return;
  const int idx = r * NCOLS + c;

  const float neffC = sN[1][t + 1];
  float sumg = 0.0f, sumv = 0.0f, cnt = 0.0f;

  if (c > 0) {                                  // left h-link (head = this node)
    const float L = len[r * (NCOLS - 1) + (c - 1)];
    sumg += (neffC - sN[1][t]) * frcp(L);
    sumv += sH[t];
    cnt += 1.0f;
  }
  if (c < NCOLS - 1) {                          // right h-link (tail = this node)
    const float L = len[r * (NCOLS - 1) + c];
    sumg += (sN[1][t + 2] - neffC) * frcp(L);
    sumv += sH[t + 1];
    cnt += 1.0f;
  }
  if (r > 0) {                                  // up v-link (head = this node)
    const float L = len[NH + (r - 1) * NCOLS + c];
    sumg += (neffC - sN[0][t + 1]) * frcp(L);
    sumv += sU[t];
    cnt += 1.0f;
  }
  if (r < NROWS - 1) {                          // down v-link (tail = this node)
    const float L = len[NH + r * NCOLS + c];
    sumg += (sN[2][t + 1] - neffC) * frcp(L);
    sumv += sD[t];
    cnt += 1.0f;
  }

  const float OPENING_COEFF  = 1.3455e-09f;
  const float CLOSURE_COEFF  = 7.11e-24f;
  const float STEP_HEIGHT    = 0.03f;
  const float INV_CUTOFF     = 1.0f / 5.74f;        // 1 / SCALE_CUTOFF
  const float INV_SEC_PER_A  = 1.0f / 31556926.0f;  // 1 / SEC_PER_A

  const float inv      = frcp(cnt);                 // cnt in {2,3,4}
  const float q        = dis[idx];
  const float gradient = sumg * inv + geo[idx];
  const float cav      = __builtin_fabsf(sumv * inv * INV_SEC_PER_A) * STEP_HEIGHT;
  const float n3       = neffC * neffC * neffC;

  float cond = (OPENING_COEFF * q * gradient + cav) *
               frcp(cav * INV_CUTOFF + CLOSURE_COEFF * n3);
  cond = (cond < 1e-6f) ? 1e-6f : cond;             // NaN passes through (jnp.where)

  // cond^1.25 = cond * sqrt(sqrt(cond));  |g|^-0.5 = rsqrt(|g|)
  const float c125 = cond * __builtin_amdgcn_sqrtf(__builtin_amdgcn_sqrtf(cond));
  const float gmh  = __builtin_amdgcn_rsqf(__builtin_fabsf(gradient));

  out[idx] = q - OPENING_COEFF * c125 * gmh * gradient;
}

extern "C" void kernel_launch(void* const* d_in, const int* in_sizes, int n_in,
                              void* d_out, int out_size, void* d_ws, size_t ws_size,
                              hipStream_t stream) {
  const float* eff    = (const float*)d_in[0];
  const float* dis    = (const float*)d_in[1];
  const float* geo    = (const float*)d_in[2];
  const float* ovb    = (const float*)d_in[3];
  const float* sv     = (const float*)d_in[4];
  const float* len    = (const float*)d_in[5];
  // d_in[6] = head, d_in[7] = tail: static raster topology, handled in closed form
  const int*   status = (const int*)d_in[8];
  float* out = (float*)d_out;

  dim3 grid((NCOLS + TILE - 1) / TILE, NROWS);
  ConduitHydrology_kernel<<<grid, TILE, 0, stream>>>(eff, dis, geo, ovb, sv, len,
                                                     status, out);
}